// GRUDecoder_56530359550211
// MI455X (gfx1250) — compile-verified
//
#include <hip/hip_runtime.h>
#include <math.h>

// ---------------------------------------------------------------------------
// GRU decoder for MI455X (gfx1250, wave32, WMMA).
// GEMMs run on V_WMMA_F32_16X16X4_F32 (exact fp32, matches reference).
// Big GEMMs stage tiles with GLOBAL_LOAD_ASYNC_TO_LDS_B128 (double-buffered,
// ASYNCcnt-pipelined) when the toolchain exposes the builtin.
// ---------------------------------------------------------------------------

typedef __attribute__((ext_vector_type(2))) float v2f;
typedef __attribute__((ext_vector_type(8))) float v8f;

// async-copy pointer types, per clang's expected builtin signature:
//   param0: int __vector_size__(16) addrspace(1)*   (global)
//   param1: int __vector_size__(16) addrspace(3)*   (LDS)
typedef int v4i_vec __attribute__((vector_size(16)));
typedef __attribute__((address_space(1))) v4i_vec* as1_v4i_ptr;
typedef __attribute__((address_space(3))) v4i_vec* as3_v4i_ptr;

#if defined(__has_builtin)
#if __has_builtin(__builtin_amdgcn_global_load_async_to_lds_b128)
#define HAVE_ASYNC_LDS 1
#endif
#endif
#ifndef HAVE_ASYNC_LDS
#define HAVE_ASYNC_LDS 0
#endif

#define BMT 64     // block tile M
#define BNT 128    // block tile N
#define BKT 32     // block tile K
#define LDAS 36    // padded LDS stride for A (float4-aligned, bank-spread)
#define LDBS 132   // padded LDS stride for B
// per-wave async instructions per staged tile: A 512/256=2, B 1024/256=4
#define ASYNC_PER_TILE 6

enum { MODE_PLAIN = 0, MODE_XPROJ = 1, MODE_ZR = 2, MODE_HH = 3, MODE_LOGITS = 4 };

__device__ __forceinline__ void async_copy16(const float* g, float* l) {
#if HAVE_ASYNC_LDS
    __builtin_amdgcn_global_load_async_to_lds_b128(
        (as1_v4i_ptr)(uintptr_t)g,
        (as3_v4i_ptr)(uintptr_t)(unsigned int)(uintptr_t)l,
        0, 0);
#else
    (void)g; (void)l;
#endif
}

__device__ __forceinline__ void async_wait_all() {
#if HAVE_ASYNC_LDS
#if __has_builtin(__builtin_amdgcn_s_wait_asynccnt)
    __builtin_amdgcn_s_wait_asynccnt(0);
#else
    asm volatile("s_wait_asynccnt 0x0" ::: "memory");
#endif
#endif
}

__device__ __forceinline__ void async_wait_tile() {  // allow next tile in flight
#if HAVE_ASYNC_LDS
#if __has_builtin(__builtin_amdgcn_s_wait_asynccnt)
    __builtin_amdgcn_s_wait_asynccnt(ASYNC_PER_TILE);
#else
    asm volatile("s_wait_asynccnt 0x6" ::: "memory");
#endif
#endif
}

// Generic WMMA f32 GEMM:  C = A[M,K] @ B[K,N]  + mode-specific epilogue.
// blockIdx.z selects {B0,C0,aux0a} vs {B1,C1,aux0b} (fuses z & r gates).
template <int MODE>
__global__ __launch_bounds__(256) void gemm_wmma(
    const float* __restrict__ A, long lda,
    const float* __restrict__ B0, const float* __restrict__ B1, int ldb,
    float* __restrict__ C0, float* __restrict__ C1, long ldc,
    int M, int N, int K,
    const float* __restrict__ aux0a,   // PLAIN: bias[N]; XPROJ/LOGITS: ctx row [B,N]; ZR/HH: x_gate
    const float* __restrict__ aux0b,   // second-gate aux for ZR
    const float* __restrict__ zbuf,    // HH: z [B,N]
    const float* __restrict__ rmul,    // HH: r [B,K]
    const float* __restrict__ hprev, long ld_hprev,  // HH: h_{t-1}
    const int* __restrict__ mask,      // ZR/HH: mask [B,T]
    int t, int T)
{
    const float* B   = (blockIdx.z == 0) ? B0 : B1;
    float* C         = (blockIdx.z == 0) ? C0 : C1;
    const float* aux = (blockIdx.z == 0) ? aux0a : aux0b;

    __shared__ float As[2][BMT][LDAS];
    __shared__ float Bs[2][BKT][LDBS];

    const int tid  = threadIdx.x;
    const int bm   = blockIdx.y * BMT;
    const int bn   = blockIdx.x * BNT;
    const int lane = tid & 31;
    const int wid  = tid >> 5;              // 8 waves
    const int wm   = (wid >> 2) * 32;       // 2x4 wave grid, 32x32 per wave
    const int wn   = (wid & 3) * 32;
    const int l16  = lane & 15;
    const int lh   = lane >> 4;             // 0/1 half-wave
    const int klo  = lh * 2;                // K sub-offset per ISA A/B layout

    v8f acc00 = {0,0,0,0,0,0,0,0};
    v8f acc01 = {0,0,0,0,0,0,0,0};
    v8f acc10 = {0,0,0,0,0,0,0,0};
    v8f acc11 = {0,0,0,0,0,0,0,0};

    // ---- 8 x (4 WMMA) inner product over one staged tile --------------------
    auto compute_tile = [&](int buf) {
#pragma unroll
        for (int kk = 0; kk < BKT; kk += 4) {
            v2f a0 = { As[buf][wm + l16][kk + klo],      As[buf][wm + l16][kk + klo + 1] };
            v2f a1 = { As[buf][wm + 16 + l16][kk + klo], As[buf][wm + 16 + l16][kk + klo + 1] };
            v2f b0 = { Bs[buf][kk + klo][wn + l16],      Bs[buf][kk + klo + 1][wn + l16] };
            v2f b1 = { Bs[buf][kk + klo][wn + 16 + l16], Bs[buf][kk + klo + 1][wn + 16 + l16] };
            acc00 = __builtin_amdgcn_wmma_f32_16x16x4_f32(false, a0, false, b0, (short)0, acc00, false, false);
            acc01 = __builtin_amdgcn_wmma_f32_16x16x4_f32(false, a0, false, b1, (short)0, acc01, false, false);
            acc10 = __builtin_amdgcn_wmma_f32_16x16x4_f32(false, a1, false, b0, (short)0, acc10, false, false);
            acc11 = __builtin_amdgcn_wmma_f32_16x16x4_f32(false, a1, false, b1, (short)0, acc11, false, false);
        }
    };

    constexpr bool ASYNC_OK =
        HAVE_ASYNC_LDS && (MODE == MODE_PLAIN || MODE == MODE_XPROJ || MODE == MODE_LOGITS);

    if constexpr (ASYNC_OK) {
        // -------- async double-buffered pipeline (raw A, raw B) --------------
        auto issue_tile = [&](int k0, int buf) {
#pragma unroll
            for (int i = tid; i < BMT * (BKT / 4); i += 256) {     // 2 per wave
                int r = i >> 3, c = (i & 7) * 4;
                int grow = bm + r; if (grow >= M) grow = M - 1;
                async_copy16(A + (long)grow * lda + k0 + c, &As[buf][r][c]);
            }
#pragma unroll
            for (int i = tid; i < BKT * (BNT / 4); i += 256) {     // 4 per wave
                int r = i >> 5, c = (i & 31) * 4;
                async_copy16(B + (long)(k0 + r) * ldb + bn + c, &Bs[buf][r][c]);
            }
        };

        const int nt = K / BKT;
        issue_tile(0, 0);
        async_wait_all();
        __syncthreads();                       // tile 0 visible everywhere
        for (int i = 0; i < nt; ++i) {
            const int cur = i & 1;
            const bool has_next = (i + 1 < nt);
            if (has_next) issue_tile((i + 1) * BKT, cur ^ 1);  // overlaps compute
            compute_tile(cur);
            if (has_next) async_wait_tile();   // tile i done; only tile i+1's copies may pend
            __syncthreads();                   // compute(i) done before its buffer is refilled
        }
    } else {
        // -------- sync staging (supports A-side elementwise transforms) ------
        for (int k0 = 0; k0 < K; k0 += BKT) {
            for (int i = tid; i < BMT * (BKT / 4); i += 256) {
                int r = i >> 3, c = (i & 7) * 4;
                int grow = bm + r; if (grow >= M) grow = M - 1;
                float4 v = *(const float4*)(A + (long)grow * lda + k0 + c);
                if (MODE == MODE_ZR || MODE == MODE_HH) {
                    float s = (t == 0) ? 0.0f : (float)mask[grow * T + (t - 1)];
                    v.x *= s; v.y *= s; v.z *= s; v.w *= s;
                    if (MODE == MODE_HH) {
                        float4 rr = *(const float4*)(rmul + (long)grow * K + k0 + c);
                        v.x *= rr.x; v.y *= rr.y; v.z *= rr.z; v.w *= rr.w;
                    }
                }
                *(float4*)&As[0][r][c] = v;
            }
            for (int i = tid; i < BKT * (BNT / 4); i += 256) {
                int r = i >> 5, c = (i & 31) * 4;
                *(float4*)&Bs[0][r][c] = *(const float4*)(B + (long)(k0 + r) * ldb + bn + c);
            }
            __syncthreads();
            compute_tile(0);
            __syncthreads();
        }
    }

    // ---- epilogue + store (C/D layout: VGPR j -> M=j+8*lh, N=l16) ----
#pragma unroll
    for (int mi = 0; mi < 2; mi++) {
#pragma unroll
        for (int ni = 0; ni < 2; ni++) {
            v8f accv = (mi == 0) ? ((ni == 0) ? acc00 : acc01)
                                 : ((ni == 0) ? acc10 : acc11);
#pragma unroll
            for (int j = 0; j < 8; j++) {
                int row = bm + wm + mi * 16 + j + lh * 8;
                int col = bn + wn + ni * 16 + l16;
                if (row < M) {
                    float v = accv[j];
                    if (MODE == MODE_PLAIN)  v += aux[col];
                    if (MODE == MODE_XPROJ)  v += aux[(long)(row / T) * N + col];
                    if (MODE == MODE_LOGITS) v += aux[(long)(row / T) * N + col];
                    if (MODE == MODE_ZR) {
                        v = v + aux[((long)row * T + t) * N + col];
                        v = 1.0f / (1.0f + __expf(-v));
                    }
                    if (MODE == MODE_HH) {
                        float hh = tanhf(v + aux[((long)row * T + t) * N + col]);
                        float s  = (t == 0) ? 0.0f : (float)mask[row * T + (t - 1)];
                        float hm = s * hprev[(long)row * ld_hprev + col];
                        float z  = zbuf[(long)row * N + col];
                        v = z * hm + (1.0f - z) * hh;
                    }
                    C[(long)row * ldc + col] = v;
                }
            }
        }
    }
}

// Build teacher-forcing shifted, masked input X[b,t,:] = target[b,t-1,:]*mask[b,t-1]
__global__ __launch_bounds__(256) void build_x(
    const float* __restrict__ tgt, const int* __restrict__ mask,
    float* __restrict__ X, int Bz, int T, int D)
{
    long i = (long)blockIdx.x * blockDim.x + threadIdx.x;
    long total = (long)Bz * T * D;
    if (i >= total) return;
    int d = (int)(i % D);
    long bt = i / D;
    int tt = (int)(bt % T);
    int b  = (int)(bt / T);
    float v = 0.0f;
    if (tt > 0)
        v = tgt[((long)b * T + (tt - 1)) * D + d] * (float)mask[b * T + (tt - 1)];
    X[i] = v;
}

// In-place row softmax over N columns, one block per row.
__global__ __launch_bounds__(256) void softmax_rows(float* __restrict__ data, int N)
{
    __shared__ float red[256];
    long base = (long)blockIdx.x * N;
    int tid = threadIdx.x;

    float mx = -3.402823466e38f;
    for (int i = tid; i < N; i += 256) mx = fmaxf(mx, data[base + i]);
    red[tid] = mx; __syncthreads();
    for (int s = 128; s > 0; s >>= 1) {
        if (tid < s) red[tid] = fmaxf(red[tid], red[tid + s]);
        __syncthreads();
    }
    float m = red[0]; __syncthreads();

    float sum = 0.0f;
    for (int i = tid; i < N; i += 256) sum += __expf(data[base + i] - m);
    red[tid] = sum; __syncthreads();
    for (int s = 128; s > 0; s >>= 1) {
        if (tid < s) red[tid] += red[tid + s];
        __syncthreads();
    }
    float inv = 1.0f / red[0];
    for (int i = tid; i < N; i += 256)
        data[base + i] = __expf(data[base + i] - m) * inv;
}

extern "C" void kernel_launch(void* const* d_in, const int* in_sizes, int n_in,
                              void* d_out, int out_size, void* d_ws, size_t ws_size,
                              hipStream_t stream) {
    (void)in_sizes; (void)n_in; (void)out_size; (void)ws_size;

    const int Bz = 32, T = 128, DIN = 512, DCTX = 512, H = 1024, V = 32000;
    const long BT = (long)Bz * T;  // 4096

    const float* target  = (const float*)d_in[0];
    const float* context = (const float*)d_in[1];
    const int*   mask    = (const int*)d_in[2];
    const float* W_z = (const float*)d_in[3],  *U_z = (const float*)d_in[4];
    const float* C_z = (const float*)d_in[5],  *b_z = (const float*)d_in[6];
    const float* W_r = (const float*)d_in[7],  *U_r = (const float*)d_in[8];
    const float* C_r = (const float*)d_in[9],  *b_r = (const float*)d_in[10];
    const float* W_h = (const float*)d_in[11], *U_h = (const float*)d_in[12];
    const float* C_h = (const float*)d_in[13], *b_h = (const float*)d_in[14];
    const float* U_y = (const float*)d_in[15], *C_y = (const float*)d_in[16];
    const float* b_y = (const float*)d_in[17];
    float* out = (float*)d_out;

    // workspace layout (floats), ~80 MB total
    float* ws   = (float*)d_ws;
    float* X    = ws;                 // BT*DIN
    float* cz   = X  + BT * DIN;      // Bz*H   (includes b_z)
    float* cr   = cz + (long)Bz * H;
    float* ch   = cr + (long)Bz * H;
    float* cy   = ch + (long)Bz * H;  // Bz*V   (includes b_y)
    float* xz   = cy + (long)Bz * V;  // BT*H
    float* xr   = xz + BT * H;
    float* xh   = xr + BT * H;
    float* hseq = xh + BT * H;        // BT*H
    float* zb   = hseq + BT * H;      // Bz*H
    float* rb   = zb + (long)Bz * H;  // Bz*H

    // 1) shifted/masked input
    {
        long total = BT * DIN;
        build_x<<<dim3((unsigned)((total + 255) / 256)), 256, 0, stream>>>(
            target, mask, X, Bz, T, DIN);
    }

    // 2) context projections (bias folded in): c* = context @ C_* + b_*
    gemm_wmma<MODE_PLAIN><<<dim3(H / BNT, 1, 1), 256, 0, stream>>>(
        context, DCTX, C_z, nullptr, H, cz, nullptr, H, Bz, H, DCTX,
        b_z, nullptr, nullptr, nullptr, nullptr, 0, nullptr, 0, T);
    gemm_wmma<MODE_PLAIN><<<dim3(H / BNT, 1, 1), 256, 0, stream>>>(
        context, DCTX, C_r, nullptr, H, cr, nullptr, H, Bz, H, DCTX,
        b_r, nullptr, nullptr, nullptr, nullptr, 0, nullptr, 0, T);
    gemm_wmma<MODE_PLAIN><<<dim3(H / BNT, 1, 1), 256, 0, stream>>>(
        context, DCTX, C_h, nullptr, H, ch, nullptr, H, Bz, H, DCTX,
        b_h, nullptr, nullptr, nullptr, nullptr, 0, nullptr, 0, T);
    gemm_wmma<MODE_PLAIN><<<dim3(V / BNT, 1, 1), 256, 0, stream>>>(
        context, DCTX, C_y, nullptr, V, cy, nullptr, V, Bz, V, DCTX,
        b_y, nullptr, nullptr, nullptr, nullptr, 0, nullptr, 0, T);

    // 3) input projections: x_g = X @ W_g + c_g[batch]
    gemm_wmma<MODE_XPROJ><<<dim3(H / BNT, (unsigned)(BT / BMT), 1), 256, 0, stream>>>(
        X, DIN, W_z, nullptr, H, xz, nullptr, H, (int)BT, H, DIN,
        cz, nullptr, nullptr, nullptr, nullptr, 0, nullptr, 0, T);
    gemm_wmma<MODE_XPROJ><<<dim3(H / BNT, (unsigned)(BT / BMT), 1), 256, 0, stream>>>(
        X, DIN, W_r, nullptr, H, xr, nullptr, H, (int)BT, H, DIN,
        cr, nullptr, nullptr, nullptr, nullptr, 0, nullptr, 0, T);
    gemm_wmma<MODE_XPROJ><<<dim3(H / BNT, (unsigned)(BT / BMT), 1), 256, 0, stream>>>(
        X, DIN, W_h, nullptr, H, xh, nullptr, H, (int)BT, H, DIN,
        ch, nullptr, nullptr, nullptr, nullptr, 0, nullptr, 0, T);

    // 4) sequential GRU scan: 2 launches per step (z,r fused; then hh+blend)
    for (int t = 0; t < T; t++) {
        const float* hprev = hseq + (long)((t > 0) ? (t - 1) : 0) * H;
        // phase 1: z = sigmoid(xz_t + hm@U_z), r = sigmoid(xr_t + hm@U_r)
        gemm_wmma<MODE_ZR><<<dim3(H / BNT, 1, 2), 256, 0, stream>>>(
            hprev, (long)T * H, U_z, U_r, H, zb, rb, H, Bz, H, H,
            xz, xr, nullptr, nullptr, nullptr, 0, mask, t, T);
        // phase 2: hh = tanh(xh_t + (r*hm)@U_h); h_t = z*hm + (1-z)*hh
        gemm_wmma<MODE_HH><<<dim3(H / BNT, 1, 1), 256, 0, stream>>>(
            hprev, (long)T * H, U_h, nullptr, H,
            hseq + (long)t * H, nullptr, (long)T * H, Bz, H, H,
            xh, nullptr, zb, rb, hprev, (long)T * H, mask, t, T);
    }

    // 5) logits straight into d_out: out = h @ U_y + cy[batch]
    gemm_wmma<MODE_LOGITS><<<dim3(V / BNT, (unsigned)(BT / BMT), 1), 256, 0, stream>>>(
        hseq, H, U_y, nullptr, V, out, nullptr, V, (int)BT, V, H,
        cy, nullptr, nullptr, nullptr, nullptr, 0, nullptr, 0, T);

    // 6) in-place softmax over V
    softmax_rows<<<dim3((unsigned)BT), 256, 0, stream>>>(out, V);
}